// NattenAttention_89661737271383
// MI455X (gfx1250) — compile-verified
//
#include <hip/hip_runtime.h>
#include <hip/hip_bf16.h>

typedef float    v2f __attribute__((ext_vector_type(2)));
typedef float    v8f __attribute__((ext_vector_type(8)));
typedef unsigned v4u __attribute__((ext_vector_type(4)));
typedef int      v4i __attribute__((ext_vector_type(4)));
typedef int      v8i __attribute__((ext_vector_type(8)));

#define HH 180
#define WW 360
#define CC 256
#define NHEAD 8
#define DHEAD 32
#define NPIX (HH * WW)          // 64800 = 32 * 2025

// ---------------------------------------------------------------------------
// TDM: DMA an A tile [32 rows x 256 fp32] from global into LDS, inserting a
// 4-DWORD pad after every 256-DWORD row -> LDS row stride = 260 floats, which
// makes the 16-lane same-k DS reads bank-conflict free. Issued by one wave,
// completion tracked with TENSORcnt.
//   group0: count=1 | lds_addr | global_addr(57b) | type=2
//   group1: data_size=4B, pad_enable, pad_interval=7 (256 dw), pad_amount=3
//           (4 dw), tensor_dim0=256, tensor_dim1=32, tile_dim0=256,
//           tile_dim1=32, tensor_dim0_stride=256, workgroup_mask=0
// ---------------------------------------------------------------------------
__device__ __forceinline__ void tdm_load_tile_A(const float* gA, void* ldsPtr) {
    const unsigned long long ga = (unsigned long long)(uintptr_t)gA;
    const unsigned lds = (unsigned)(uintptr_t)ldsPtr;   // low 32b = LDS byte offset

    v4u g0;
    g0[0] = 1u;                                          // count=1 (valid), user mode
    g0[1] = lds;                                         // lds_addr
    g0[2] = (unsigned)(ga & 0xffffffffu);                // global_addr[31:0]
    g0[3] = (unsigned)((ga >> 32) & 0x01ffffffu)         // global_addr[56:32]
          | 0x80000000u;                                 // type=2 ("image")

    v8i g1;
    g1[0] = (2 << 16)      // data_size = 4 bytes
          | (1 << 20)      // pad_enable
          | (7 << 22)      // pad_interval: 256 DWORDs
          | (3 << 25);     // pad_amount: 4 DWORDs
    g1[1] = (int)(256u << 16);   // tensor_dim0 = 256   ([31:16] = low half)
    g1[2] = (int)(32u  << 16);   // tensor_dim0 hi = 0, tensor_dim1 = 32
    g1[3] = (int)(256u << 16);   // tensor_dim1 hi = 0, tile_dim0 = 256
    g1[4] = 32;                  // tile_dim1 = 32, tile_dim2 = 0
    g1[5] = 256;                 // tensor_dim0_stride = 256 (low 32)
    g1[6] = 0;                   // stride hi, tensor_dim1_stride low (unused, 2D)
    g1[7] = 0;

    v4i z4 = {0, 0, 0, 0};       // groups 2/3 unused (2-D tensor)
#if defined(__clang_major__) && (__clang_major__ >= 23)
    v8i z8 = {0, 0, 0, 0, 0, 0, 0, 0};
    __builtin_amdgcn_tensor_load_to_lds(g0, g1, z4, z4, z8, 0);
#else
    __builtin_amdgcn_tensor_load_to_lds(g0, g1, z4, z4, 0);
#endif
}

// ---------------------------------------------------------------------------
// GEMM: C[M x N] = A[M x 256] * B[256 x N] + bias[N]   (fp32 WMMA 16x16x4)
// block = 256 threads = 8 waves. Block tile: 32(M) x 128(N); wave tile 32x16
// (two 16x16 accumulators sharing one B fragment -> half the B traffic/FLOP).
// N is a compile-time constant so every B/C offset folds into the 24-bit
// immediate-offset field of global_load/store (no per-iteration address VALU),
// and the fully unrolled K loop removes the prefetch guard branch.
// grid = (N/128, M/32). M % 32 == 0, N % 128 == 0 (guaranteed by caller).
// ---------------------------------------------------------------------------
template <int N>
__global__ __launch_bounds__(256)
void gemm_wmma_f32(const float* __restrict__ A, const float* __restrict__ B,
                   const float* __restrict__ bias, float* __restrict__ C)
{
    constexpr int K = 256;
    constexpr int LDSS = 260;               // 4*m mod 64 distinct -> no bank conflicts
    __shared__ float sA[32 * LDSS];

    const int m_base = blockIdx.y * 32;
    const int wave   = threadIdx.x >> 5;
    const int lane   = threadIdx.x & 31;
    const int half   = lane >> 4;           // 0 or 1
    const int t      = lane & 15;
    const int n      = blockIdx.x * 128 + wave * 16 + t;

    // One wave DMAs the whole A tile via the Tensor Data Mover.
    if (threadIdx.x < 32) {
        tdm_load_tile_A(A + (size_t)m_base * K, (void*)sA);
        __builtin_amdgcn_s_wait_tensorcnt(0);
    }
    __syncthreads();

    const float bn = bias[n];
    v8f acc0, acc1;
    #pragma unroll
    for (int r = 0; r < 8; ++r) { acc0[r] = bn; acc1[r] = bn; }

    const float* sArow0 = &sA[t * LDSS];          // rows m_base + 0..15
    const float* sArow1 = &sA[(16 + t) * LDSS];   // rows m_base + 16..31
    const float* Bcol   = B + n;

    #pragma unroll
    for (int k0 = 0; k0 < K; k0 += 4) {
        const int kk = k0 + 2 * half;
        v2f b, a0, a1;
        b.x  = Bcol[(size_t)kk * N];              // const offsets -> imm field
        b.y  = Bcol[(size_t)(kk + 1) * N];
        a0.x = sArow0[kk];
        a0.y = sArow0[kk + 1];
        a1.x = sArow1[kk];
        a1.y = sArow1[kk + 1];
        if (k0 + 16 < K)                          // compile-time under full unroll
            __builtin_prefetch(&Bcol[(size_t)(kk + 16) * N], 0, 1);
        acc0 = __builtin_amdgcn_wmma_f32_16x16x4_f32(
                   false, a0, false, b, (short)0, acc0, false, false);
        acc1 = __builtin_amdgcn_wmma_f32_16x16x4_f32(
                   false, a1, false, b, (short)0, acc1, false, false);
    }

    float* Crow0 = C + (size_t)(m_base +      8 * half) * N + n;
    float* Crow1 = C + (size_t)(m_base + 16 + 8 * half) * N + n;
    #pragma unroll
    for (int r = 0; r < 8; ++r) {
        Crow0[(size_t)r * N] = acc0[r];
        Crow1[(size_t)r * N] = acc1[r];
    }
}

// ---------------------------------------------------------------------------
// Neighborhood attention core. One thread = one (pixel, head).
// qkv layout per pixel: [3][NH][32] (cols of the QKV GEMM output).
// Height: clamped 7-window (NATTEN). Width: circular (reference wraps W), so
// start_w is never clamped and the rpb column base is constant (= 3).
// ---------------------------------------------------------------------------
__global__ __launch_bounds__(256)
void natten_attn(const float* __restrict__ qkv, const float* __restrict__ rpb,
                 float* __restrict__ out)
{
    const int tid = blockIdx.x * blockDim.x + threadIdx.x;
    if (tid >= NPIX * NHEAD) return;
    const int head = tid & 7;
    const int pix  = tid >> 3;
    const int h = pix / WW;
    const int w = pix - h * WW;

    const float scale = 0.17677669529663687f;   // 32^-0.5

    int start_h = h - 3;
    if (start_h < 0) start_h = 0;
    if (start_h > HH - 7) start_h = HH - 7;
    const int relh0 = start_h - h + 6;          // rpb row base

    const float* qp = qkv + (size_t)pix * 768 + head * 32;
    float4 q[8];
    #pragma unroll
    for (int i = 0; i < 8; ++i) q[i] = reinterpret_cast<const float4*>(qp)[i];

    float s[49];
    float mx = -1e30f;
    const float* rp = rpb + head * 169;         // [13][13]

    #pragma unroll
    for (int oi = 0; oi < 7; ++oi) {
        const int kh = start_h + oi;
        #pragma unroll
        for (int oj = 0; oj < 7; ++oj) {
            int kw = w + oj - 3;
            if (kw < 0) kw += WW; else if (kw >= WW) kw -= WW;
            const float4* k4 = reinterpret_cast<const float4*>(
                qkv + (size_t)(kh * WW + kw) * 768 + 256 + head * 32);
            float acc = 0.f;
            #pragma unroll
            for (int i = 0; i < 8; ++i) {
                float4 kv = k4[i];
                acc += q[i].x * kv.x + q[i].y * kv.y + q[i].z * kv.z + q[i].w * kv.w;
            }
            const float sc = acc * scale + rp[(relh0 + oi) * 13 + (oj + 3)];
            s[oi * 7 + oj] = sc;
            mx = fmaxf(mx, sc);
        }
    }

    float4 o[8];
    #pragma unroll
    for (int i = 0; i < 8; ++i) o[i] = make_float4(0.f, 0.f, 0.f, 0.f);
    float sum = 0.f;

    #pragma unroll
    for (int oi = 0; oi < 7; ++oi) {
        const int kh = start_h + oi;
        #pragma unroll
        for (int oj = 0; oj < 7; ++oj) {
            int kw = w + oj - 3;
            if (kw < 0) kw += WW; else if (kw >= WW) kw -= WW;
            const float p = __expf(s[oi * 7 + oj] - mx);
            sum += p;
            const float4* v4 = reinterpret_cast<const float4*>(
                qkv + (size_t)(kh * WW + kw) * 768 + 512 + head * 32);
            #pragma unroll
            for (int i = 0; i < 8; ++i) {
                float4 vv = v4[i];
                o[i].x += p * vv.x; o[i].y += p * vv.y;
                o[i].z += p * vv.z; o[i].w += p * vv.w;
            }
        }
    }

    const float inv = 1.f / sum;
    float4* op = reinterpret_cast<float4*>(out + (size_t)pix * 256 + head * 32);
    #pragma unroll
    for (int i = 0; i < 8; ++i) {
        o[i].x *= inv; o[i].y *= inv; o[i].z *= inv; o[i].w *= inv;
        op[i] = o[i];
    }
}

// ---------------------------------------------------------------------------
// Launch: qkv GEMM -> neighborhood attention -> output projection GEMM.
// Scratch: qkv [64800 x 768] fp32 (199 MB) then attn [64800 x 256] fp32 (66 MB).
// ---------------------------------------------------------------------------
extern "C" void kernel_launch(void* const* d_in, const int* in_sizes, int n_in,
                              void* d_out, int out_size, void* d_ws, size_t ws_size,
                              hipStream_t stream) {
    (void)in_sizes; (void)n_in; (void)out_size; (void)ws_size;

    const float* x      = (const float*)d_in[0];   // [1,180,360,256]
    const float* w_qkv  = (const float*)d_in[1];   // [256, 768]
    const float* b_qkv  = (const float*)d_in[2];   // [768]
    const float* rpb    = (const float*)d_in[3];   // [8,13,13]
    const float* w_proj = (const float*)d_in[4];   // [256, 256]
    const float* b_proj = (const float*)d_in[5];   // [256]
    float* out = (float*)d_out;                    // [1,180,360,256]

    float* qkv  = (float*)d_ws;                                               // 64800*768
    float* attn = (float*)((char*)d_ws + (size_t)NPIX * 768 * sizeof(float)); // 64800*256

    dim3 blk(256);
    // QKV projection: [64800 x 256] x [256 x 768]
    gemm_wmma_f32<768><<<dim3(768 / 128, NPIX / 32), blk, 0, stream>>>(x, w_qkv, b_qkv, qkv);
    // Neighborhood attention: 64800*8 (pixel, head) threads
    natten_attn<<<dim3((NPIX * NHEAD) / 256), blk, 0, stream>>>(qkv, rpb, attn);
    // Output projection: [64800 x 256] x [256 x 256]
    gemm_wmma_f32<256><<<dim3(256 / 128, NPIX / 32), blk, 0, stream>>>(attn, w_proj, b_proj, out);
}